// PANScorer_14044543057998
// MI455X (gfx1250) — compile-verified
//
#include <hip/hip_runtime.h>

typedef __attribute__((ext_vector_type(2))) float v2f;
typedef __attribute__((ext_vector_type(8))) float v8f;

#define N_NODES 100000
#define FEAT 256
#define FEAT_V4 (FEAT / 4)          // 64 float4 per row
#define ROWS 16                     // rows per wave tile
#define TPAD 260                    // padded LDS row stride (260 % 64 == 4 -> no bank conflicts)
#define TPAD_V4 (TPAD / 4)          // 65

// ---------------------------------------------------------------------------
// gfx1250 async global->LDS copy (ASYNCcnt-tracked, bypasses VGPR staging).
// VDST operand = per-lane LDS byte address; per ISA, low 32 bits of a generic
// pointer to __shared__ are the wave-relative LDS address.
// ---------------------------------------------------------------------------
__device__ __forceinline__ unsigned lds_addr_of(const void* p) {
  return (unsigned)(unsigned long long)p;
}
__device__ __forceinline__ void async_ld_b128(unsigned lds_off, const void* gptr) {
  asm volatile("global_load_async_to_lds_b128 %0, %1, off"
               :: "v"(lds_off), "v"(gptr) : "memory");
}
__device__ __forceinline__ void wait_async0() {
  asm volatile("s_wait_asynccnt 0x0" ::: "memory");
}

// ---------------------------------------------------------------------------
// Kernel 1: score2[c] += edge_weight  (segment sum via L2-resident f32 atomics)
// ---------------------------------------------------------------------------
__global__ __launch_bounds__(256) void pan_edge_segsum(
    const float* __restrict__ ew, const long long* __restrict__ col,
    float* __restrict__ score2, int n_edges) {
  int i = blockIdx.x * blockDim.x + threadIdx.x;
  if (i < n_edges) {
    float w = ew[i];
    int c = (int)col[i];
    atomicAdd(&score2[c], w);   // global_atomic_add_f32, 400KB target lives in L2
  }
}

// ---------------------------------------------------------------------------
// Kernel 2: per 16-row tile (one wave32):
//   LDS <- x tile via GLOBAL_LOAD_ASYNC_TO_LDS_B128 (single HBM pass over x)
//   score1 via V_WMMA_F32_16X16X4_F32 accumulation over K=256
//   score = sigmoid(b0*score1 + b1*score2); out = tile * score
// ---------------------------------------------------------------------------
__global__ __launch_bounds__(32) void pan_score_scale(
    const float* __restrict__ x, const float* __restrict__ p,
    const float* __restrict__ beta, const float* __restrict__ score2,
    float* __restrict__ out_scaled, float* __restrict__ out_score) {
  __shared__ float tile[ROWS * TPAD];   // 16.25 KB, padded rows
  __shared__ float pl[FEAT];
  __shared__ float s1[ROWS];
  __shared__ float sc[ROWS];

  const int tid = threadIdx.x;                       // 0..31 (wave32)
  const long long row_base = (long long)blockIdx.x * ROWS;

  // ---- async global -> LDS copy of 16x256 f32 tile (b128, LDS rows padded) ----
  const float4* xg = (const float4*)(x + row_base * FEAT);
  const unsigned tile_lds = lds_addr_of(tile);
#pragma unroll
  for (int it = 0; it < (ROWS * FEAT_V4) / 32; ++it) {   // 32 async b128 per lane
    int idx = tid + 32 * it;                             // 0..1023
    int r = idx >> 6;                                    // row
    int c = idx & 63;                                    // float4 column
    async_ld_b128(tile_lds + (unsigned)(r * TPAD_V4 + c) * 16u, xg + idx);
  }
  // ---- p (1KB) -> LDS, async ----
  const float4* pg = (const float4*)p;
  const unsigned pl_lds = lds_addr_of(pl);
#pragma unroll
  for (int it = 0; it < FEAT_V4 / 32; ++it) {
    int idx = tid + 32 * it;
    async_ld_b128(pl_lds + (unsigned)idx * 16u, pg + idx);
  }
  wait_async0();
  __syncthreads();

  // ---- score1 via WMMA f32 16x16x4, accumulate over K ----
  // A layout (16x4 f32): lanes 0-15 -> M=lane, VGPR0=K0,VGPR1=K1;
  //                      lanes 16-31 -> M=lane-16, VGPR0=K2,VGPR1=K3.
  // B (4x16): p[k] replicated across all N columns -> same lane pattern.
  const int half = tid >> 4;        // 0: K-pair {0,1}, 1: K-pair {2,3}
  const int m    = tid & 15;        // row within tile
  const int koff = 2 * half;
  v8f acc = {};
#pragma unroll 8
  for (int kb = 0; kb < FEAT; kb += 4) {
    v2f a, b;
    a.x = tile[m * TPAD + kb + koff];
    a.y = tile[m * TPAD + kb + koff + 1];
    b.x = pl[kb + koff];
    b.y = pl[kb + koff + 1];
    acc = __builtin_amdgcn_wmma_f32_16x16x4_f32(
        /*neg_a=*/false, a, /*neg_b=*/false, b,
        /*c_mod=*/(short)0, acc, /*reuse_a=*/false, /*reuse_b=*/false);
  }

  // Every column of D equals score1[row]. C/D layout: lane n (n<16) VGPR v
  // holds (M=v, N=n); lane 16+n VGPR v holds (M=v+8, N=n). Spill column N=0.
  if (m == 0) {
#pragma unroll
    for (int v = 0; v < 8; ++v) s1[half * 8 + v] = acc[v];
  }
  __syncthreads();

  // ---- score = sigmoid(b0*score1 + b1*score2); emit score output ----
  if (tid < ROWS) {
    float z = beta[0] * s1[tid] + beta[1] * score2[row_base + tid];
    float s = __builtin_amdgcn_rcpf(1.0f + __expf(-z));   // v_rcp_f32 sigmoid
    sc[tid] = s;
    out_score[row_base + tid] = s;
  }
  __syncthreads();

  // ---- scale tile from LDS, b128 stores ----
  const float4* t4 = (const float4*)tile;
  float4* og = (float4*)(out_scaled + row_base * FEAT);
#pragma unroll
  for (int it = 0; it < (ROWS * FEAT_V4) / 32; ++it) {
    int idx = tid + 32 * it;
    int r = idx >> 6;
    int c = idx & 63;
    float s = sc[r];
    float4 v = t4[r * TPAD_V4 + c];
    v.x *= s; v.y *= s; v.z *= s; v.w *= s;
    og[idx] = v;
  }
}

// ---------------------------------------------------------------------------
extern "C" void kernel_launch(void* const* d_in, const int* in_sizes, int n_in,
                              void* d_out, int out_size, void* d_ws, size_t ws_size,
                              hipStream_t stream) {
  const float*     x    = (const float*)d_in[0];
  const float*     ew   = (const float*)d_in[1];
  const long long* col  = (const long long*)d_in[2];
  const float*     p    = (const float*)d_in[3];
  const float*     beta = (const float*)d_in[4];

  float* out       = (float*)d_out;
  float* out_score = out + (size_t)N_NODES * FEAT;   // tuple: [x*score | score]
  float* score2    = (float*)d_ws;                   // 400 KB scratch
  const int n_edges = in_sizes[1];

  hipMemsetAsync(score2, 0, N_NODES * sizeof(float), stream);
  pan_edge_segsum<<<(n_edges + 255) / 256, 256, 0, stream>>>(ew, col, score2, n_edges);
  pan_score_scale<<<N_NODES / ROWS, 32, 0, stream>>>(x, p, beta, score2, out, out_score);
}